// SA_sub_block_12309376270384
// MI455X (gfx1250) — compile-verified
//
#include <hip/hip_runtime.h>
#include <cstdint>

typedef float v2f __attribute__((ext_vector_type(2)));
typedef float v8f __attribute__((ext_vector_type(8)));

#define C_   64
#define B_   8
#define H_   128
#define W_   128
#define HW   16384            // H_*W_
#define NTOT (B_ * HW)        // 131072 spatial columns
#define NELT (B_ * C_ * HW)   // 8388608 output elements

// ---------------------------------------------------------------------------
// Kernel A: fused QKV GEMM.  D(192xN) = [Wq;Wk;Wv](192x64) * X(64xN) + bias.
// One wave32 computes all 12 M-tiles (192 rows) for one 16-column N-tile
// using V_WMMA_F32_16X16X4_F32 (exact fp32 MACs, matches reference numerics).
// ---------------------------------------------------------------------------
__global__ __launch_bounds__(128)
void qkv_gemm(const float* __restrict__ x,
              const float* __restrict__ Wq, const float* __restrict__ bq,
              const float* __restrict__ Wk, const float* __restrict__ bk,
              const float* __restrict__ Wv, const float* __restrict__ bv,
              float* __restrict__ Qo, float* __restrict__ Ko, float* __restrict__ Vo)
{
    const int lane = threadIdx.x & 31;
    const int wave = threadIdx.x >> 5;
    const int n0   = blockIdx.x * 64 + wave * 16;   // global column base (16-col tile)
    const int b    = n0 / HW;                       // 16384 % 64 == 0 -> no batch straddle
    const int p    = n0 - b * HW;
    const int n    = lane & 15;                     // column within tile
    const int hi   = lane >> 4;                     // 0 or 1 (lane half)
    const int koff = hi * 2;                        // A/B K-pair offset per ISA layout

    const float* xb = x + (size_t)b * C_ * HW + p + n;

    const float* Wmat[3] = {Wq, Wk, Wv};
    const float* bias[3] = {bq, bk, bv};

    // 12 accumulator tiles (Q rows 0..63, K rows 64..127, V rows 128..191),
    // initialized with the per-channel bias (C/D layout: VGPR r -> M = r + hi*8).
    v8f acc[12];
#pragma unroll
    for (int t = 0; t < 12; ++t) {
        const float* bp  = bias[t >> 2];
        const int    ch0 = (t & 3) * 16 + hi * 8;
#pragma unroll
        for (int r = 0; r < 8; ++r) acc[t][r] = bp[ch0 + r];
    }

    // K = 64 reduction in 16 steps of 4.
    for (int ck = 0; ck < 64; ck += 4) {
        // B-frag (4x16): lane holds X[ck+koff+{0,1}, n]
        v2f bfrag;
        bfrag.x = xb[(size_t)(ck + koff)     * HW];
        bfrag.y = xb[(size_t)(ck + koff + 1) * HW];
#pragma unroll
        for (int t = 0; t < 12; ++t) {
            const float* wm  = Wmat[t >> 2];
            const int    row = (t & 3) * 16 + (lane & 15);
            // A-frag (16x4): lane holds W[row, ck+koff+{0,1}] (contiguous, 8B aligned)
            v2f afrag = *(const v2f*)(wm + row * 64 + ck + koff);
            acc[t] = __builtin_amdgcn_wmma_f32_16x16x4_f32(
                false, afrag, false, bfrag, (short)0, acc[t], false, false);
        }
    }

    float* Omat[3] = {Qo, Ko, Vo};
#pragma unroll
    for (int t = 0; t < 12; ++t) {
        float*    op  = Omat[t >> 2] + (size_t)b * C_ * HW + p + n;
        const int ch0 = (t & 3) * 16 + hi * 8;
#pragma unroll
        for (int r = 0; r < 8; ++r)
            op[(size_t)(ch0 + r) * HW] = acc[t][r];
    }
}

// ---------------------------------------------------------------------------
// Kernel B: 3x3 windowed attention. OOB taps use bk[c]/bv[c] (conv of zero pad
// is bias). Writes attention output to d_out and accumulates per-channel
// sum / sum-of-squares for batchnorm (each block lives in one (b,c) plane).
// ---------------------------------------------------------------------------
__global__ __launch_bounds__(256)
void attn_kernel(const float* __restrict__ Q, const float* __restrict__ Kp,
                 const float* __restrict__ Vp,
                 const float* __restrict__ bk, const float* __restrict__ bv,
                 const float* __restrict__ rel_h, const float* __restrict__ rel_w,
                 float* __restrict__ out, float* __restrict__ sums)
{
    __shared__ float s1[256];
    __shared__ float s2[256];
    const int idx = blockIdx.x * 256 + threadIdx.x;
    const int w = idx & (W_ - 1);
    const int h = (idx >> 7) & (H_ - 1);
    const int c = (idx >> 14) & (C_ - 1);

    const float q = Q[idx];
    const float* rp = (c < 32) ? (rel_h + c * 9) : (rel_w + (c - 32) * 9);
    const float bkc = bk[c], bvc = bv[c];

    float s[9], vv[9];
    float m = -3.4e38f;
#pragma unroll
    for (int i = 0; i < 3; ++i) {
#pragma unroll
        for (int j = 0; j < 3; ++j) {
            const int  t  = i * 3 + j;
            const int  hh = h + i - 1, ww = w + j - 1;
            const bool in = ((unsigned)hh < (unsigned)H_) && ((unsigned)ww < (unsigned)W_);
            const int  off = (i - 1) * W_ + (j - 1);
            const float kv = in ? Kp[idx + off] : bkc;
            vv[t] = in ? Vp[idx + off] : bvc;
            s[t]  = q * (kv + rp[t]);
            m = fmaxf(m, s[t]);
        }
    }
    float esum = 0.f, accv = 0.f;
#pragma unroll
    for (int t = 0; t < 9; ++t) {
        const float e = __expf(s[t] - m);
        esum += e;
        accv += e * vv[t];
    }
    const float o = accv / esum;
    out[idx] = o;

    // block reduction (all 256 threads share channel c: 256 | HW plane)
    s1[threadIdx.x] = o;
    s2[threadIdx.x] = o * o;
    __syncthreads();
    for (int st = 128; st > 0; st >>= 1) {
        if (threadIdx.x < st) {
            s1[threadIdx.x] += s1[threadIdx.x + st];
            s2[threadIdx.x] += s2[threadIdx.x + st];
        }
        __syncthreads();
    }
    if (threadIdx.x == 0) {
        atomicAdd(&sums[c],      s1[0]);
        atomicAdd(&sums[64 + c], s2[0]);
    }
}

// ---------------------------------------------------------------------------
// Kernel C: per-channel batchnorm stats -> affine scale/shift.
// ---------------------------------------------------------------------------
__global__ void bn_stats(const float* __restrict__ sums,
                         const float* __restrict__ gamma,
                         const float* __restrict__ beta,
                         float* __restrict__ ss)
{
    const int c = threadIdx.x;
    if (c < C_) {
        const float inv_n  = 1.0f / (float)(B_ * HW);
        const float mean   = sums[c] * inv_n;
        const float var    = sums[64 + c] * inv_n - mean * mean;
        const float invstd = rsqrtf(var + 1e-5f);
        const float sc     = gamma[c] * invstd;
        ss[c]      = sc;
        ss[64 + c] = beta[c] - mean * sc;
    }
}

// ---------------------------------------------------------------------------
// Kernel D: y = relu(out*scale + shift + x), in place on d_out.
// ---------------------------------------------------------------------------
__global__ __launch_bounds__(256)
void bn_apply(float* __restrict__ out, const float* __restrict__ x,
              const float* __restrict__ ss)
{
    const int idx = blockIdx.x * 256 + threadIdx.x;
    const int c   = (idx >> 14) & (C_ - 1);
    const float y = out[idx] * ss[c] + ss[64 + c] + x[idx];
    out[idx] = fmaxf(y, 0.f);
}

// ---------------------------------------------------------------------------
extern "C" void kernel_launch(void* const* d_in, const int* in_sizes, int n_in,
                              void* d_out, int out_size, void* d_ws, size_t ws_size,
                              hipStream_t stream)
{
    (void)in_sizes; (void)n_in; (void)out_size; (void)ws_size;
    const float* x     = (const float*)d_in[0];
    const float* Wq    = (const float*)d_in[1];
    const float* bq    = (const float*)d_in[2];
    const float* Wk    = (const float*)d_in[3];
    const float* bk    = (const float*)d_in[4];
    const float* Wv    = (const float*)d_in[5];
    const float* bv    = (const float*)d_in[6];
    const float* rel_h = (const float*)d_in[7];
    const float* rel_w = (const float*)d_in[8];
    const float* gamma = (const float*)d_in[9];
    const float* beta  = (const float*)d_in[10];
    float* out = (float*)d_out;

    // workspace layout: Q | K | V | sums(128) | scale+shift(128)
    float* wsf  = (float*)d_ws;
    float* Qb   = wsf;
    float* Kb   = wsf + (size_t)NELT;
    float* Vb   = wsf + 2 * (size_t)NELT;
    float* sums = wsf + 3 * (size_t)NELT;
    float* ss   = sums + 128;

    hipMemsetAsync(sums, 0, 128 * sizeof(float), stream);

    qkv_gemm<<<NTOT / 64, 128, 0, stream>>>(x, Wq, bq, Wk, bk, Wv, bv, Qb, Kb, Vb);
    attn_kernel<<<NELT / 256, 256, 0, stream>>>(Qb, Kb, Vb, bk, bv, rel_h, rel_w, out, sums);
    bn_stats<<<1, 64, 0, stream>>>(sums, gamma, beta, ss);
    bn_apply<<<NELT / 256, 256, 0, stream>>>(out, x, ss);
}